// DifferentiableMPCController_13211319402802
// MI455X (gfx1250) — compile-verified
//
#include <hip/hip_runtime.h>
#include <math.h>
#include <stdint.h>

#define NXd 12
#define NUd 4
#define Td  256
#define DTd 0.05f
#define REGd 1e-6f
#define MAX_ITERd 5

typedef float v2f __attribute__((ext_vector_type(2)));
typedef float v8f __attribute__((ext_vector_type(8)));

// V_WMMA_F32_16X16X4_F32 : D(16x16 f32) = A(16x4 f32) * B(4x16 f32) + C
__device__ __forceinline__ v8f wmma4(v2f a, v2f b, v8f c) {
  return __builtin_amdgcn_wmma_f32_16x16x4_f32(
      /*neg_a=*/false, a, /*neg_b=*/false, b,
      /*c_mod=*/(short)0, c, /*reuse_a=*/false, /*reuse_b=*/false);
}

// Async global->LDS DMA of one 16x16 f32 tile (row stride 17 floats).
// Each lane moves 8 contiguous floats (one half-row) as two B128 transfers.
// INST_OFFSET is added to BOTH the LDS and global addresses (ISA 08 §4.4),
// so one base pair covers both 16-byte halves. Tracked with ASYNCcnt.
__device__ __forceinline__ void async_tile_load(float (*dst)[17], const float* src, int l) {
  const int row = l >> 1, col = (l & 1) * 8;
  uint32_t lds0 = (uint32_t)(uintptr_t)(&dst[row][col]);   // low 32 bits = LDS byte offset
  const float* g0 = src + l * 8;
  asm volatile("global_load_async_to_lds_b128 %0, %1, off"
               :: "v"(lds0), "v"(g0) : "memory");
  asm volatile("global_load_async_to_lds_b128 %0, %1, off offset:16"
               :: "v"(lds0), "v"(g0) : "memory");
}
// ASYNCcnt completions are in-order: <=2 outstanding => previous tile landed.
__device__ __forceinline__ void wait_async2() {
  asm volatile("s_wait_asynccnt 0x2" ::: "memory");
}
__device__ __forceinline__ void wait_async0() {
  asm volatile("s_wait_asynccnt 0x0" ::: "memory");
}

struct SharedMem {
  float Ct[2][16][17]; // double-buffered C[t] tile (Ct[0] also reused for C_final)
  float Vt[16][17];    // value Hessian V (padded, zeros outside 12x12)
  float Ft[16][17];    // F = [A | B] (rows 12..15 zero)
  float Gt[16][17];    // G = F^T V
  float Qt[16][17];    // Q = C + F^T V F (+ REG on uu diag)
  float WxL[12][12];
  float WuL[12][4];
  float vvec[16];      // value gradient v (entries 12..15 zero)
  float qlin[16];      // Qx | Qu
  float tau[16];
  float xu[16];        // [x(t) | u(t)]
  float KtL[4][12];    // K[t]
  float ktL[4];        // k[t]
  float xcur[12];
  float unew[4];
  float xs[8][12];     // per-alpha states
  float us[8][4];      // per-alpha controls
  float taus[8][16];
  float red[32];
  float acost[8];
  float scal[4];
};

__global__ __launch_bounds__(32)
void ilqr_kernel(const float* __restrict__ x0g,
                 const float* __restrict__ Uinit,
                 const float* __restrict__ Cg,
                 const float* __restrict__ cg,
                 const float* __restrict__ Cfg,
                 const float* __restrict__ cfg,
                 const float* __restrict__ xrefg,
                 const float* __restrict__ urefg,
                 const float* __restrict__ Wxg,
                 const float* __restrict__ Wug,
                 float* __restrict__ outX,
                 float* __restrict__ outU,
                 float* __restrict__ ws)
{
  __shared__ SharedMem S;
  const int b  = blockIdx.x;
  const int l  = threadIdx.x;
  const int lo = l & 15;
  const int hi = l >> 4;

  const float* Cb   = Cg   + (size_t)b * Td * 256;
  const float* cb   = cg   + (size_t)b * Td * 16;
  const float* Cf   = Cfg  + (size_t)b * 256;
  const float* cf   = cfg  + (size_t)b * 16;
  const float* xref = xrefg + (size_t)b * (Td + 1) * NXd;
  const float* uref = urefg + (size_t)b * Td * NUd;
  const float* x0p  = x0g  + (size_t)b * NXd;
  const float* U0   = Uinit + (size_t)b * Td * NUd;

  // workspace layout per batch (floats): X[2][3084], U[2][1024], K[12288], k[1024]
  float* wsb = ws + (size_t)b * 21528;
  float* Xbuf0 = wsb;
  float* Xbuf1 = wsb + 3084;
  float* Ubuf0 = wsb + 6168;
  float* Ubuf1 = wsb + 6168 + 1024;
  float* Kw = wsb + 8216;
  float* kw = wsb + 20504;

  for (int idx = l; idx < 144; idx += 32) S.WxL[idx / 12][idx % 12] = Wxg[idx];
  for (int idx = l; idx < 48;  idx += 32) S.WuL[idx / 4][idx % 4]  = Wug[idx];
  __syncthreads();

  int cur = 0;
  float best;

  // ========== initial rollout with U_init + total cost ==========
  {
    for (int idx = l; idx < Td * NUd; idx += 32) Ubuf0[idx] = U0[idx];
    if (l < NXd) { S.xcur[l] = x0p[l]; Xbuf0[l] = x0p[l]; }
    float costacc = 0.f;
    async_tile_load(S.Ct[0], Cb, l);                 // prime tile t=0
    __syncthreads();
    for (int t = 0; t < Td; ++t) {
      const int p = t & 1;
      if (t + 1 < Td) async_tile_load(S.Ct[p ^ 1], Cb + (size_t)(t + 1) * 256, l);
      if (l < NUd) S.unew[l] = U0[t * NUd + l];
      __syncthreads();
      if (l < 16) S.tau[l] = (l < NXd) ? (S.xcur[l] - xref[t * NXd + l])
                                       : (S.unew[l - NXd] - uref[t * NUd + l - NXd]);
      if (t + 1 < Td) wait_async2(); else wait_async0();   // tile t landed
      __syncthreads();
      if (l < 16) {
        float rd = 0.f;
        #pragma unroll
        for (int j = 0; j < 16; ++j) rd += S.Ct[p][l][j] * S.tau[j];
        costacc += (0.5f * rd + cb[t * 16 + l]) * S.tau[l];
      }
      float xn = 0.f;
      if (l < NXd) {
        float z = 0.f;
        #pragma unroll
        for (int j = 0; j < NXd; ++j) z += S.WxL[l][j] * S.xcur[j];
        #pragma unroll
        for (int j = 0; j < NUd; ++j) z += S.WuL[l][j] * S.unew[j];
        xn = S.xcur[l] + DTd * tanhf(z);
      }
      __syncthreads();
      if (l < NXd) { S.xcur[l] = xn; Xbuf0[(t + 1) * NXd + l] = xn; }
      __syncthreads();
    }
    // terminal cost
    #pragma unroll
    for (int q = 0; q < 8; ++q) { int idx = l * 8 + q; S.Ct[0][idx >> 4][idx & 15] = Cf[idx]; }
    __syncthreads();
    if (l < 16) S.tau[l] = (l < NXd) ? (S.xcur[l] - xref[Td * NXd + l]) : 0.f;
    __syncthreads();
    if (l < 16) {
      float rd = 0.f;
      #pragma unroll
      for (int j = 0; j < 16; ++j) rd += S.Ct[0][l][j] * S.tau[j];
      costacc += (0.5f * rd + cf[l]) * S.tau[l];
    }
    S.red[l] = costacc;
    __syncthreads();
    if (l == 0) {
      float tot = 0.f;
      for (int q = 0; q < 16; ++q) tot += S.red[q];
      S.scal[0] = tot;
    }
    __syncthreads();
    best = S.scal[0];
    __syncthreads();
  }

  // ========== iLQR iterations ==========
  for (int iter = 0; iter < MAX_ITERd; ++iter) {
    const float* Xc = cur ? Xbuf1 : Xbuf0;
    const float* Uc = cur ? Ubuf1 : Ubuf0;

    // ----- terminal V, v -----
    #pragma unroll
    for (int q = 0; q < 8; ++q) { int idx = l * 8 + q; S.Ct[0][idx >> 4][idx & 15] = Cf[idx]; }
    __syncthreads();
    if (l < 16) S.tau[l] = (l < NXd) ? (Xc[Td * NXd + l] - xref[Td * NXd + l]) : 0.f;
    __syncthreads();
    if (l < 16) {
      float rd = 0.f;
      #pragma unroll
      for (int j = 0; j < 16; ++j) rd += S.Ct[0][l][j] * S.tau[j];
      S.vvec[l] = (l < NXd) ? (rd + cf[l]) : 0.f;
    }
    #pragma unroll
    for (int q = 0; q < 8; ++q) {
      int idx = l * 8 + q; int m = idx >> 4, n2 = idx & 15;
      S.Vt[m][n2] = (m < NXd && n2 < NXd) ? Cf[idx] : 0.f;
    }
    async_tile_load(S.Ct[(Td - 1) & 1], Cb + (size_t)(Td - 1) * 256, l);  // prime t=T-1
    __syncthreads();

    // ----- backward Riccati scan -----
    for (int t = Td - 1; t >= 0; --t) {
      const int p = t & 1;
      if (t > 0) async_tile_load(S.Ct[p ^ 1], Cb + (size_t)(t - 1) * 256, l);
      if (l < 16) S.xu[l] = (l < NXd) ? Xc[t * NXd + l] : Uc[t * NUd + (l - NXd)];
      __syncthreads();
      if (l < 16) S.tau[l] = S.xu[l] - ((l < NXd) ? xref[t * NXd + l] : uref[t * NUd + l - NXd]);
      // build F = [A | B] (rows 12..15 zero)
      if (l < 16) {
        if (l < NXd) {
          float z = 0.f;
          #pragma unroll
          for (int j = 0; j < NXd; ++j) z += S.WxL[l][j] * S.xu[j];
          #pragma unroll
          for (int j = 0; j < NUd; ++j) z += S.WuL[l][j] * S.xu[NXd + j];
          float h = tanhf(z);
          float gg = DTd * (1.f - h * h);
          #pragma unroll
          for (int j = 0; j < 16; ++j)
            S.Ft[l][j] = (j < NXd) ? ((l == j ? 1.f : 0.f) + gg * S.WxL[l][j])
                                   : (gg * S.WuL[l][j - NXd]);
        } else {
          #pragma unroll
          for (int j = 0; j < 16; ++j) S.Ft[l][j] = 0.f;
        }
      }
      if (t > 0) wait_async2(); else wait_async0();   // tile t landed
      __syncthreads();
      // qlin = C tau + c + F^T v
      if (l < 16) {
        float rd = 0.f;
        #pragma unroll
        for (int j = 0; j < 16; ++j) rd += S.Ct[p][l][j] * S.tau[j];
        rd += cb[t * 16 + l];
        #pragma unroll
        for (int j = 0; j < NXd; ++j) rd += S.Ft[j][l] * S.vvec[j];
        S.qlin[l] = rd;
      }
      // G = F^T * V  (three chained 16x16x4 WMMAs over K=12)
      v8f g8 = {0.f, 0.f, 0.f, 0.f, 0.f, 0.f, 0.f, 0.f};
      #pragma unroll
      for (int k0 = 0; k0 < NXd; k0 += 4) {
        v2f a, bb;
        a.x  = S.Ft[k0 + 2 * hi][lo];     a.y  = S.Ft[k0 + 2 * hi + 1][lo];   // (F^T)[m][k] = F[k][m]
        bb.x = S.Vt[k0 + 2 * hi][lo];     bb.y = S.Vt[k0 + 2 * hi + 1][lo];
        g8 = wmma4(a, bb, g8);
      }
      #pragma unroll
      for (int j = 0; j < 8; ++j) S.Gt[j + 8 * hi][lo] = g8[j];
      __syncthreads();
      // Q = C + G * F
      v8f q8;
      #pragma unroll
      for (int j = 0; j < 8; ++j) q8[j] = S.Ct[p][j + 8 * hi][lo];
      #pragma unroll
      for (int k0 = 0; k0 < NXd; k0 += 4) {
        v2f a, bb;
        a.x  = S.Gt[lo][k0 + 2 * hi];     a.y  = S.Gt[lo][k0 + 2 * hi + 1];
        bb.x = S.Ft[k0 + 2 * hi][lo];     bb.y = S.Ft[k0 + 2 * hi + 1][lo];
        q8 = wmma4(a, bb, q8);
      }
      #pragma unroll
      for (int j = 0; j < 8; ++j) {
        int m = j + 8 * hi;
        float val = q8[j];
        if (m >= NXd && m == lo) val += REGd;   // + REG * I_uu
        S.Qt[m][lo] = val;
      }
      __syncthreads();
      // 4x4 SPD solve: columns 0..11 -> -K, column 12 -> -k (per-lane RHS, replicated LU)
      {
        float M00 = S.Qt[12][12], M01 = S.Qt[12][13], M02 = S.Qt[12][14], M03 = S.Qt[12][15];
        float M11 = S.Qt[13][13], M12 = S.Qt[13][14], M13 = S.Qt[13][15];
        float M22 = S.Qt[14][14], M23 = S.Qt[14][15];
        float M33 = S.Qt[15][15];
        float L10 = S.Qt[13][12], L20 = S.Qt[14][12], L30 = S.Qt[15][12];
        float L21 = S.Qt[14][13], L31 = S.Qt[15][13];
        float L32 = S.Qt[15][14];
        int col = (l < 13) ? l : 12;
        float r0, r1, r2, r3;
        if (col < 12) { r0 = S.Qt[12][col]; r1 = S.Qt[13][col]; r2 = S.Qt[14][col]; r3 = S.Qt[15][col]; }
        else          { r0 = S.qlin[12];    r1 = S.qlin[13];    r2 = S.qlin[14];    r3 = S.qlin[15]; }
        float f;
        f = L10 / M00; M11 -= f * M01; M12 -= f * M02; M13 -= f * M03; r1 -= f * r0;
        f = L20 / M00; L21 -= f * M01; M22 -= f * M02; M23 -= f * M03; r2 -= f * r0;
        f = L30 / M00; L31 -= f * M01; L32 -= f * M02; M33 -= f * M03; r3 -= f * r0;
        f = L21 / M11; M22 -= f * M12; M23 -= f * M13; r2 -= f * r1;
        f = L31 / M11; L32 -= f * M12; M33 -= f * M13; r3 -= f * r1;
        f = L32 / M22; M33 -= f * M23; r3 -= f * r2;
        float s3 = r3 / M33;
        float s2 = (r2 - M23 * s3) / M22;
        float s1 = (r1 - M12 * s2 - M13 * s3) / M11;
        float s0 = (r0 - M01 * s1 - M02 * s2 - M03 * s3) / M00;
        if (l < 12) {
          S.KtL[0][l] = -s0; S.KtL[1][l] = -s1; S.KtL[2][l] = -s2; S.KtL[3][l] = -s3;
          Kw[t * 48 + 0 * 12 + l] = -s0; Kw[t * 48 + 1 * 12 + l] = -s1;
          Kw[t * 48 + 2 * 12 + l] = -s2; Kw[t * 48 + 3 * 12 + l] = -s3;
        } else if (l == 12) {
          S.ktL[0] = -s0; S.ktL[1] = -s1; S.ktL[2] = -s2; S.ktL[3] = -s3;
          kw[t * 4 + 0] = -s0; kw[t * 4 + 1] = -s1; kw[t * 4 + 2] = -s2; kw[t * 4 + 3] = -s3;
        }
      }
      __syncthreads();
      // V' = Qxx + Qux^T K (single 16x16x4 WMMA), v' = Qx + Qux^T k
      {
        v2f a, bb;
        a.x  = (lo < NXd) ? S.Qt[12 + 2 * hi][lo]     : 0.f;   // (Qux^T)[m][kk]
        a.y  = (lo < NXd) ? S.Qt[12 + 2 * hi + 1][lo] : 0.f;
        bb.x = (lo < NXd) ? S.KtL[2 * hi][lo]         : 0.f;   // K[kk][n]
        bb.y = (lo < NXd) ? S.KtL[2 * hi + 1][lo]     : 0.f;
        v8f vn8;
        #pragma unroll
        for (int j = 0; j < 8; ++j) vn8[j] = S.Qt[j + 8 * hi][lo];
        vn8 = wmma4(a, bb, vn8);
        float newv = 0.f;
        if (l < NXd) {
          newv = S.qlin[l];
          #pragma unroll
          for (int kk = 0; kk < NUd; ++kk) newv += S.Qt[12 + kk][l] * S.ktL[kk];
        }
        __syncthreads();
        #pragma unroll
        for (int j = 0; j < 8; ++j) {
          int m = j + 8 * hi;
          S.Vt[m][lo] = (m < NXd && lo < NXd) ? vn8[j] : 0.f;
        }
        if (l < 16) S.vvec[l] = (l < NXd) ? newv : 0.f;
      }
      __syncthreads();
    }

    // ----- line search: all 8 alphas in parallel (4 lanes each) -----
    {
      const int a = l >> 2, r = l & 3;
      const float alpha = exp2f(-(float)a);
      #pragma unroll
      for (int q = 0; q < 3; ++q) S.xs[a][r * 3 + q] = x0p[r * 3 + q];
      float costacc = 0.f;
      async_tile_load(S.Ct[0], Cb, l);               // prime tile t=0
      __syncthreads();
      for (int t = 0; t < Td; ++t) {
        const int p = t & 1;
        if (t + 1 < Td) async_tile_load(S.Ct[p ^ 1], Cb + (size_t)(t + 1) * 256, l);
        for (int idx = l; idx < 52; idx += 32) {
          if (idx < 48) S.KtL[idx / 12][idx % 12] = Kw[t * 48 + idx];
          else          S.ktL[idx - 48] = kw[t * 4 + (idx - 48)];
        }
        __syncthreads();
        float uval = Uc[t * NUd + r] + alpha * S.ktL[r];
        #pragma unroll
        for (int j = 0; j < NXd; ++j) uval += S.KtL[r][j] * (S.xs[a][j] - Xc[t * NXd + j]);
        S.us[a][r] = uval;
        __syncthreads();
        #pragma unroll
        for (int q = 0; q < 4; ++q) {
          int e = 4 * r + q;
          S.taus[a][e] = (e < NXd) ? (S.xs[a][e] - xref[t * NXd + e])
                                   : (S.us[a][e - NXd] - uref[t * NUd + e - NXd]);
        }
        if (t + 1 < Td) wait_async2(); else wait_async0();   // tile t landed
        __syncthreads();
        #pragma unroll
        for (int q = 0; q < 4; ++q) {
          int e = 4 * r + q;
          float rd = 0.f;
          #pragma unroll
          for (int j = 0; j < 16; ++j) rd += S.Ct[p][e][j] * S.taus[a][j];
          costacc += (0.5f * rd + cb[t * 16 + e]) * S.taus[a][e];
        }
        float xn[3];
        #pragma unroll
        for (int q = 0; q < 3; ++q) {
          int i = 3 * r + q;
          float z = 0.f;
          #pragma unroll
          for (int j = 0; j < NXd; ++j) z += S.WxL[i][j] * S.xs[a][j];
          #pragma unroll
          for (int j = 0; j < NUd; ++j) z += S.WuL[i][j] * S.us[a][j];
          xn[q] = S.xs[a][i] + DTd * tanhf(z);
        }
        __syncthreads();
        #pragma unroll
        for (int q = 0; q < 3; ++q) S.xs[a][3 * r + q] = xn[q];
        __syncthreads();
      }
      // terminal cost + per-alpha reduction
      #pragma unroll
      for (int q = 0; q < 8; ++q) { int idx = l * 8 + q; S.Ct[0][idx >> 4][idx & 15] = Cf[idx]; }
      S.red[l] = costacc;
      __syncthreads();
      if (r == 0) {
        float tot = S.red[4 * a] + S.red[4 * a + 1] + S.red[4 * a + 2] + S.red[4 * a + 3];
        float tn[NXd];
        #pragma unroll
        for (int j = 0; j < NXd; ++j) tn[j] = S.xs[a][j] - xref[Td * NXd + j];
        float fin = 0.f;
        #pragma unroll
        for (int i = 0; i < NXd; ++i) {
          float rd = 0.f;
          #pragma unroll
          for (int j = 0; j < NXd; ++j) rd += S.Ct[0][i][j] * tn[j];
          fin += (0.5f * rd + cf[i]) * tn[i];
        }
        S.acost[a] = tot + fin;
      }
      __syncthreads();
      if (l == 0) {
        float bc = S.acost[0]; int bi = 0;
        for (int q = 1; q < 8; ++q) if (S.acost[q] < bc) { bc = S.acost[q]; bi = q; }
        S.scal[0] = bc; S.scal[1] = (float)bi;
      }
      __syncthreads();
    }

    // ----- accept (ping-pong) and re-roll best alpha storing trajectory -----
    float candcost = S.scal[0];
    int besta = (int)S.scal[1];
    __syncthreads();
    if (candcost < best) {
      const float al = exp2f(-(float)besta);
      int nxt = cur ^ 1;
      float* Xn = nxt ? Xbuf1 : Xbuf0;
      float* Un = nxt ? Ubuf1 : Ubuf0;
      if (l < NXd) { S.xcur[l] = x0p[l]; Xn[l] = x0p[l]; }
      __syncthreads();
      for (int t = 0; t < Td; ++t) {
        for (int idx = l; idx < 52; idx += 32) {
          if (idx < 48) S.KtL[idx / 12][idx % 12] = Kw[t * 48 + idx];
          else          S.ktL[idx - 48] = kw[t * 4 + (idx - 48)];
        }
        __syncthreads();
        if (l < NUd) {
          float uval = Uc[t * NUd + l] + al * S.ktL[l];
          #pragma unroll
          for (int j = 0; j < NXd; ++j) uval += S.KtL[l][j] * (S.xcur[j] - Xc[t * NXd + j]);
          S.unew[l] = uval;
          Un[t * NUd + l] = uval;
        }
        __syncthreads();
        float xn = 0.f;
        if (l < NXd) {
          float z = 0.f;
          #pragma unroll
          for (int j = 0; j < NXd; ++j) z += S.WxL[l][j] * S.xcur[j];
          #pragma unroll
          for (int j = 0; j < NUd; ++j) z += S.WuL[l][j] * S.unew[j];
          xn = S.xcur[l] + DTd * tanhf(z);
        }
        __syncthreads();
        if (l < NXd) { S.xcur[l] = xn; Xn[(t + 1) * NXd + l] = xn; }
        __syncthreads();
      }
      cur = nxt;
      best = candcost;
    }
    __syncthreads();
  }

  // ----- write outputs -----
  const float* Xf = cur ? Xbuf1 : Xbuf0;
  const float* Uf = cur ? Ubuf1 : Ubuf0;
  float* oX = outX + (size_t)b * 3084;
  float* oU = outU + (size_t)b * 1024;
  for (int idx = l; idx < 3084; idx += 32) oX[idx] = Xf[idx];
  for (int idx = l; idx < 1024; idx += 32) oU[idx] = Uf[idx];
}

extern "C" void kernel_launch(void* const* d_in, const int* in_sizes, int n_in,
                              void* d_out, int out_size, void* d_ws, size_t ws_size,
                              hipStream_t stream) {
  const float* x0   = (const float*)d_in[0];
  const float* Uin  = (const float*)d_in[1];
  const float* C    = (const float*)d_in[2];
  const float* c    = (const float*)d_in[3];
  const float* Cfin = (const float*)d_in[4];
  const float* cfin = (const float*)d_in[5];
  const float* xref = (const float*)d_in[6];
  const float* uref = (const float*)d_in[7];
  const float* Wx   = (const float*)d_in[8];
  const float* Wu   = (const float*)d_in[9];

  const int Bn = in_sizes[0] / NXd;          // batch size (512)
  float* outX = (float*)d_out;
  float* outU = outX + (size_t)Bn * (Td + 1) * NXd;
  float* ws = (float*)d_ws;

  ilqr_kernel<<<Bn, 32, 0, stream>>>(x0, Uin, C, c, Cfin, cfin, xref, uref, Wx, Wu,
                                     outX, outU, ws);
  (void)n_in; (void)out_size; (void)ws_size;
}